// TopKPooling_89223650607314
// MI455X (gfx1250) — compile-verified
//
#include <hip/hip_runtime.h>
#include <cfloat>
#include <cstdint>

// Problem constants (match reference setup_inputs)
#define TOPK 8
#define DCH 256
#define LSEQ 8192
#define NB 32

#define TILE_ROWS 16                         // rows (l values) per LDS tile
#define TILE_FLOATS (TILE_ROWS * DCH)        // 4096 floats = 16 KB
#define CHUNKS_PER_THREAD 4                  // (16KB / 16B) / 256 threads

// ---------------------------------------------------------------------------
// Async global->LDS staging (CDNA5 ASYNCcnt path). Each call copies 4 x 16 = 64 bytes
// per lane from global memory into LDS. Tracked by ASYNCcnt, waited with
// s_wait_asynccnt. LDS byte address = low 32 bits of the generic pointer.
// ---------------------------------------------------------------------------
__device__ __forceinline__ void async_tile_issue(const float* __restrict__ gsrc,
                                                 float* lbuf, int tid) {
#pragma unroll
    for (int c = 0; c < CHUNKS_PER_THREAD; ++c) {
        const int chunk = tid + c * 256;                 // 16B chunk index
        const unsigned lds = (unsigned)(unsigned long long)(const void*)&lbuf[chunk * 4];
        const float* g = gsrc + chunk * 4;
        asm volatile("global_load_async_to_lds_b128 %0, %1, off"
                     :: "v"(lds), "v"(g)
                     : "memory");
    }
}

__device__ __forceinline__ void wait_async_le(int n) {
    if (n == 0) asm volatile("s_wait_asynccnt 0" ::: "memory");
    else        asm volatile("s_wait_asynccnt 4" ::: "memory");
}

// Branchless sorted-insert helper: tk[] sorted descending, tk[7] replaced,
// single bubble pass restores order.
__device__ __forceinline__ void topk_insert(float (&tk)[TOPK], float v) {
    if (v > tk[TOPK - 1]) {
        tk[TOPK - 1] = v;
#pragma unroll
        for (int j = TOPK - 1; j >= 1; --j) {
            const float hi = fmaxf(tk[j - 1], tk[j]);
            const float lo = fminf(tk[j - 1], tk[j]);
            tk[j - 1] = hi;
            tk[j]     = lo;
        }
    }
}

// ---------------------------------------------------------------------------
// Phase 1: per-segment partial top-8 per channel.
// grid = (S, NB), block = 256 threads (8 wave32). Thread d owns channel d.
// Streams the segment through a double-buffered 16KB LDS tile using async
// copies; LDS reads are bank-conflict-free (lane d -> bank d).
// part layout: part[((b*S + s)*TOPK + k)*DCH + d], each (b,s,d) list sorted
// descending. Masked rows contribute -FLT_MAX (== finfo(f32).min fill).
// ---------------------------------------------------------------------------
__global__ __launch_bounds__(256) void topk_partial_kernel(
    const float* __restrict__ x, const int* __restrict__ xmask,
    float* __restrict__ part, int S, int SEG) {
    __shared__ float buf[2][TILE_FLOATS];

    const int tid = threadIdx.x;
    const int s   = blockIdx.x;
    const int b   = blockIdx.y;
    const int d   = tid;
    const int ntiles = SEG / TILE_ROWS;

    const float* gx  = x + ((long long)b * LSEQ + (long long)s * SEG) * DCH;
    const int*  mrow = xmask + (long long)b * LSEQ + (long long)s * SEG;

    float tk[TOPK];
#pragma unroll
    for (int k = 0; k < TOPK; ++k) tk[k] = -FLT_MAX;

    // Prime the pipeline.
    async_tile_issue(gx, buf[0], tid);

    for (int t = 0; t < ntiles; ++t) {
        const bool more = (t + 1) < ntiles;
        if (more) async_tile_issue(gx + (size_t)(t + 1) * TILE_FLOATS, buf[(t + 1) & 1], tid);
        wait_async_le(more ? 4 : 0);     // tile t complete; next tile in flight
        __syncthreads();

        const float* tb = buf[t & 1];
        const int lbase = t * TILE_ROWS;
#pragma unroll
        for (int r = 0; r < TILE_ROWS; ++r) {
            const int m = mrow[lbase + r];           // wave-uniform address
            float v = tb[r * DCH + d];               // ds_load, conflict-free
            v = m ? -FLT_MAX : v;
            topk_insert(tk, v);
        }
        __syncthreads();                 // readers done before buffer reuse
    }

    float* pout = part + (((long long)b * S + s) * TOPK) * DCH + d;
#pragma unroll
    for (int k = 0; k < TOPK; ++k) pout[(long long)k * DCH] = tk[k];
}

// ---------------------------------------------------------------------------
// Phase 2: merge S partial sorted-8 lists per (b, d) -> final top-8.
// grid = NB blocks, 256 threads; coalesced reads (unit stride across lanes, DCH across lists)
// and coalesced writes in the reference's [B, k*D] layout.
// ---------------------------------------------------------------------------
__global__ __launch_bounds__(256) void topk_merge_kernel(
    const float* __restrict__ part, float* __restrict__ out, int S) {
    const int d = threadIdx.x;
    const int b = blockIdx.x;

    float tk[TOPK];
#pragma unroll
    for (int k = 0; k < TOPK; ++k) tk[k] = -FLT_MAX;

    const float* p = part + ((long long)b * S * TOPK) * DCH + d;
    const int total = S * TOPK;
    for (int i = 0; i < total; ++i) {
        topk_insert(tk, p[(long long)i * DCH]);
    }

    float* o = out + (long long)b * (TOPK * DCH) + d;
#pragma unroll
    for (int k = 0; k < TOPK; ++k) o[k * DCH] = tk[k];
}

// ---------------------------------------------------------------------------
// Launch: pick segment count S (power of two <= 32) so partials fit in d_ws.
// For S == 1 the partial layout equals the output layout, so phase 1 can
// write d_out directly and the merge is skipped (also the no-workspace path).
// ---------------------------------------------------------------------------
extern "C" void kernel_launch(void* const* d_in, const int* in_sizes, int n_in,
                              void* d_out, int out_size, void* d_ws, size_t ws_size,
                              hipStream_t stream) {
    (void)in_sizes; (void)n_in; (void)out_size;
    const float* x  = (const float*)d_in[0];
    const int*   xm = (const int*)d_in[1];
    float* out = (float*)d_out;

    int S = 32;
    while (S > 1 && (size_t)NB * S * TOPK * DCH * sizeof(float) > ws_size) S >>= 1;

    const bool have_ws = ((size_t)NB * S * TOPK * DCH * sizeof(float) <= ws_size);
    if (!have_ws) S = 1;

    const int SEG = LSEQ / S;
    if (S == 1) {
        // Partial layout == output layout for S == 1: write d_out directly.
        topk_partial_kernel<<<dim3(1, NB), 256, 0, stream>>>(x, xm, out, 1, SEG);
    } else {
        float* part = (float*)d_ws;
        topk_partial_kernel<<<dim3(S, NB), 256, 0, stream>>>(x, xm, part, S, SEG);
        topk_merge_kernel<<<NB, 256, 0, stream>>>(part, out, S);
    }
}